// VectorQuantizer_24146306138443
// MI455X (gfx1250) — compile-verified
//
#include <hip/hip_runtime.h>

typedef float v2f __attribute__((ext_vector_type(2)));
typedef float v8f __attribute__((ext_vector_type(8)));

#define NUM_VOCAB 16384
#define NZ        8192      // 8*32*32 z-vectors
#define HW        1024      // 32*32
#define CHW       4096      // 4*1024
#define NWAVES    512       // NZ / 16

// One wave handles 16 z-vectors against the whole codebook using
// v_wmma_f32_16x16x4_f32 (K=4 == DIM_EMBED).
__global__ __launch_bounds__(256) void vq_argmin_kernel(
    const float* __restrict__ x,        // [8,4,32,32]
    const float* __restrict__ cb,       // [16384,4]
    float* __restrict__ out,            // d_out[0 .. 32768)
    float* __restrict__ idx_out,        // d_out + 32769, 8192 floats
    float* __restrict__ ws)             // per-wave loss partials (512 floats)
{
    const int tid  = blockIdx.x * blockDim.x + threadIdx.x;
    const int wave = tid >> 5;
    const int lane = threadIdx.x & 31;
    const int h    = lane >> 4;     // which 16-lane half
    const int nl   = lane & 15;     // A-row / B-column within tile
    const int zbase = wave * 16;
    const bool hi  = (h != 0);

    // ---- A fragment: row m = nl, K components {2h, 2h+1} (ISA 16x4 f32 layout)
    int zi  = zbase + nl;
    int bb  = zi >> 10;             // batch
    int rem = zi & 1023;            // hh*32 + w
    const float* xz = x + bb * CHW + rem;
    v2f a;
    a.x = xz[(2 * h)     * HW];
    a.y = xz[(2 * h + 1) * HW];

    v8f czero = {};

    float best[8];
    int   bidx[8];
#pragma unroll
    for (int r = 0; r < 8; ++r) { best[r] = 3.402823e38f; bidx[r] = 0; }

    // Each lane loads the FULL float4 row of code (base+nl): gives both the
    // B-fragment half (selected by h) and the complete ||e_n||^2 locally,
    // with no cross-lane shuffle in the hot loop.
    const float4* cbl = (const float4*)cb + nl;

#pragma unroll 2
    for (int chunk = 0; chunk < NUM_VOCAB / 16; ++chunk) {
        float4 e4 = cbl[chunk * 16];                 // 16 codes * 16B per chunk
        float ee = (e4.x * e4.x + e4.y * e4.y) + (e4.z * e4.z + e4.w * e4.w);
        v2f b;
        b.x = hi ? e4.z : e4.x;
        b.y = hi ? e4.w : e4.y;

        v8f acc = __builtin_amdgcn_wmma_f32_16x16x4_f32(
            false, a, false, b, (short)0, czero, false, false);

        int code = (chunk << 4) + nl;
#pragma unroll
        for (int r = 0; r < 8; ++r) {
            // score = ||e||^2 - 2*dot  (||z||^2 dropped: constant per row)
            float s = fmaf(-2.0f, acc[r], ee);
            if (s < best[r]) { best[r] = s; bidx[r] = code; }
        }
    }

    // ---- argmin reduce over the 16 columns (xor stays within each half) ----
#pragma unroll
    for (int r = 0; r < 8; ++r) {
#pragma unroll
        for (int off = 1; off < 16; off <<= 1) {
            float os = __shfl_xor(best[r], off, 32);
            int   oi = __shfl_xor(bidx[r], off, 32);
            // branchless tie-break: smaller score, then smaller index
            bool take = (os < best[r]) | ((os == best[r]) & (oi < bidx[r]));
            best[r] = take ? os : best[r];
            bidx[r] = take ? oi : bidx[r];
        }
    }

    // ---- writeback: lane 0 -> rows 0..7, lane 16 -> rows 8..15 ----
    float lsum = 0.0f;
    if (nl == 0) {
        const float4* cb4 = (const float4*)cb;
#pragma unroll
        for (int r = 0; r < 8; ++r) {
            int m    = r + 8 * h;
            int z    = zbase + m;
            int bb2  = z >> 10;
            int rem2 = z & 1023;
            const float* xr   = x   + bb2 * CHW + rem2;
            float*       orow = out + bb2 * CHW + rem2;
            float4 e = cb4[bidx[r]];
            float zc, d;
            zc = xr[0 * HW]; d = e.x - zc; orow[0 * HW] = zc + d; lsum += d * d;
            zc = xr[1 * HW]; d = e.y - zc; orow[1 * HW] = zc + d; lsum += d * d;
            zc = xr[2 * HW]; d = e.z - zc; orow[2 * HW] = zc + d; lsum += d * d;
            zc = xr[3 * HW]; d = e.w - zc; orow[3 * HW] = zc + d; lsum += d * d;
            idx_out[z] = (float)bidx[r];
        }
    }
    lsum += __shfl_xor(lsum, 16, 32);   // combine lane0 + lane16 partials
    if (lane == 0) ws[wave] = lsum;
}

// Deterministic fixed-order tree reduction of the 512 per-wave partials.
__global__ __launch_bounds__(512) void vq_loss_reduce(
    const float* __restrict__ ws, float* __restrict__ loss_out)
{
    __shared__ float sm[NWAVES];
    int t = threadIdx.x;
    sm[t] = ws[t];
    __syncthreads();
    for (int s = NWAVES / 2; s > 0; s >>= 1) {
        if (t < s) sm[t] += sm[t + s];
        __syncthreads();
    }
    if (t == 0) {
        // loss = mean(d^2) + 0.25*mean(d^2) = 1.25 * sum / 32768
        loss_out[0] = 1.25f * sm[0] * (1.0f / 32768.0f);
    }
}

extern "C" void kernel_launch(void* const* d_in, const int* in_sizes, int n_in,
                              void* d_out, int out_size, void* d_ws, size_t ws_size,
                              hipStream_t stream) {
    const float* x  = (const float*)d_in[0];   // [8,4,32,32]
    const float* cb = (const float*)d_in[1];   // [16384,4]
    float* out  = (float*)d_out;               // 32768 floats
    float* loss = out + 32768;                 // 1 float
    float* idxo = out + 32769;                 // 8192 floats
    float* ws   = (float*)d_ws;                // 512 floats of scratch

    // 64 blocks * 8 waves = 512 waves, 16 z-vectors per wave
    vq_argmin_kernel<<<64, 256, 0, stream>>>(x, cb, out, idxo, ws);
    vq_loss_reduce<<<1, NWAVES, 0, stream>>>(ws, loss);
}